// ProbAttention_79362405695799
// MI455X (gfx1250) — compile-verified
//
#include <hip/hip_runtime.h>

// ---------------------------------------------------------------------------
// ProbSparse attention (Informer) for MI455X / gfx1250, wave32 + WMMA fp32
// + async global->LDS staging (ASYNCcnt) with double buffering.
// Shapes fixed by the reference: H=8, B=32, L=4096, D=64, sample_k=n_top=45.
// ---------------------------------------------------------------------------

#define HB_TOTAL   256          // H*B
#define SEQ_L      4096
#define DIM_D      64
#define N_SAMPLE   45
#define N_TOP      45
#define N_TOP_PAD  48           // 3 tiles of 16
#define SCALE      0.125f       // 1/sqrt(64)
#define KB_COUNT   (SEQ_L / 16) // 256 key blocks
#define LDS_PITCH  65           // 64 + 1 pad column (bank-conflict-free col reads)

typedef __attribute__((ext_vector_type(2))) float v2f;
typedef __attribute__((ext_vector_type(8))) float v8f;

// Low 32 bits of a flat LDS pointer == LDS byte offset (LDS aperture layout).
static __device__ __forceinline__ uint32_t lds_byte_off(const void* p) {
    return (uint32_t)(uintptr_t)p;
}

static __device__ __forceinline__ void wait_asynccnt0() {
    asm volatile("s_wait_asynccnt 0x0" ::: "memory");
}

// One B128 async copy: 16 bytes global -> 16 bytes LDS (per active lane).
static __device__ __forceinline__ void async_b128(uint32_t lds_off, uint32_t voff,
                                                  uint64_t sbase) {
    asm volatile("global_load_async_to_lds_b128 %0, %1, %2"
                 :: "v"(lds_off), "v"(voff), "s"(sbase)
                 : "memory");
}

// Stage one 16x64 fp32 block (rows kb*16..kb*16+15) into a padded LDS buffer.
// 256 float4 groups, 96 threads -> 3 rounds.
static __device__ __forceinline__ void issue_k_stage(uint32_t kbuf_off, uint64_t kbase,
                                                     int kb, int tid) {
    for (int it = 0; it < 3; ++it) {
        const int g = tid + it * 96;
        if (g < 256) {
            const int r = g >> 4, c = (g & 15) * 4;
            const uint32_t lo = kbuf_off + (uint32_t)((r * LDS_PITCH + c) * 4);
            const uint32_t vo = (uint32_t)(((kb * 16 + r) * DIM_D + c) * 4);
            async_b128(lo, vo, kbase);
        }
    }
}

static __device__ __forceinline__ void issue_kv_stage(uint32_t kbuf_off, uint64_t kbase,
                                                      uint32_t vbuf_off, uint64_t vbase,
                                                      int kb, int tid) {
    for (int it = 0; it < 3; ++it) {
        const int g = tid + it * 96;
        if (g < 256) {
            const int r = g >> 4, c = (g & 15) * 4;
            const uint32_t ldsrel = (uint32_t)((r * LDS_PITCH + c) * 4);
            const uint32_t vo = (uint32_t)(((kb * 16 + r) * DIM_D + c) * 4);
            async_b128(kbuf_off + ldsrel, vo, kbase);
            async_b128(vbuf_off + ldsrel, vo, vbase);
        }
    }
}

// ---------------------------------------------------------------------------
// Kernel 1: sampled QK scores -> sparsity measure M = max_s - sum_s / L
// One wave32 per query; lane holds float2 of the 64-dim row.
// ---------------------------------------------------------------------------
__global__ __launch_bounds__(256) void sample_m_kernel(
    const float* __restrict__ q, const float* __restrict__ k,
    const int* __restrict__ sidx, float* __restrict__ M)
{
    const int gwave = (int)((blockIdx.x * blockDim.x + threadIdx.x) >> 5);
    const int lane  = threadIdx.x & 31;
    // gwave == hb * L + i  (q layout [H,B,L,D] flattened)
    const int hb = gwave >> 12;
    const int i  = gwave & (SEQ_L - 1);

    const float* qrow = q + (size_t)gwave * DIM_D;
    const float  qx = qrow[2 * lane];
    const float  qy = qrow[2 * lane + 1];
    const float* kb = k + (size_t)hb * SEQ_L * DIM_D;
    const int*   sr = sidx + (size_t)i * N_SAMPLE;

    float mx = -INFINITY, sm = 0.0f;
    for (int s = 0; s < N_SAMPLE; ++s) {
        const float* kr = kb + (size_t)sr[s] * DIM_D;
        float p = qx * kr[2 * lane] + qy * kr[2 * lane + 1];
        p += __shfl_xor(p, 1, 32);
        p += __shfl_xor(p, 2, 32);
        p += __shfl_xor(p, 4, 32);
        p += __shfl_xor(p, 8, 32);
        p += __shfl_xor(p, 16, 32);
        mx = fmaxf(mx, p);
        sm += p;
    }
    if (lane == 0) M[gwave] = mx - sm * (1.0f / (float)SEQ_L);
}

// ---------------------------------------------------------------------------
// Kernel 2: top-45 indices of M per (h,b). One block per (h,b).
// ---------------------------------------------------------------------------
__global__ __launch_bounds__(256) void topk_kernel(
    const float* __restrict__ M, int* __restrict__ m_top)
{
    __shared__ float vals[SEQ_L];
    __shared__ float rv[256];
    __shared__ int   ri[256];
    const int hb = blockIdx.x, tid = threadIdx.x;
    const float* Mb = M + (size_t)hb * SEQ_L;
    for (int e = tid; e < SEQ_L; e += 256) vals[e] = Mb[e];
    __syncthreads();

    for (int t = 0; t < N_TOP; ++t) {
        float bv = -INFINITY; int bi = 0;
        for (int e = tid; e < SEQ_L; e += 256) {
            float v = vals[e];
            if (v > bv) { bv = v; bi = e; }
        }
        rv[tid] = bv; ri[tid] = bi;
        __syncthreads();
        for (int s = 128; s > 0; s >>= 1) {
            if (tid < s) {
                if (rv[tid + s] > rv[tid]) { rv[tid] = rv[tid + s]; ri[tid] = ri[tid + s]; }
            }
            __syncthreads();
        }
        if (tid == 0) {
            m_top[hb * N_TOP_PAD + t] = ri[0];
            vals[ri[0]] = -INFINITY;
        }
        __syncthreads();
    }
    if (tid < N_TOP_PAD - N_TOP) m_top[hb * N_TOP_PAD + N_TOP + tid] = 0;
}

// ---------------------------------------------------------------------------
// Kernel 3: vmean[h,b,d] = mean over L of v. One block per (h,b).
// ---------------------------------------------------------------------------
__global__ __launch_bounds__(256) void vmean_kernel(
    const float* __restrict__ v, float* __restrict__ vmean)
{
    __shared__ float part[4][DIM_D];
    const int hb = blockIdx.x, tid = threadIdx.x;
    const int d = tid & 63, g = tid >> 6;
    const float* vb = v + (size_t)hb * SEQ_L * DIM_D;
    float s = 0.0f;
    for (int r = g; r < SEQ_L; r += 4) s += vb[(size_t)r * DIM_D + d];
    part[g][d] = s;
    __syncthreads();
    if (g == 0)
        vmean[hb * DIM_D + d] =
            (part[0][d] + part[1][d] + part[2][d] + part[3][d]) * (1.0f / (float)SEQ_L);
}

// ---------------------------------------------------------------------------
// Kernel 4: broadcast fill ctx[h,b,i,:] = vmean[h,b,:]  (float4 / B128 stores)
// ---------------------------------------------------------------------------
__global__ __launch_bounds__(256) void fill_kernel(
    const float* __restrict__ vmean, float4* __restrict__ out4)
{
    const size_t i = (size_t)blockIdx.x * blockDim.x + threadIdx.x;  // float4 idx
    const size_t e = i * 4;                                          // element idx
    const int hb = (int)(e >> 18);      // / (L*D)
    const int d  = (int)(e & 63);       // multiple of 4
    const float* vm = vmean + hb * DIM_D + d;
    float4 val; val.x = vm[0]; val.y = vm[1]; val.z = vm[2]; val.w = vm[3];
    out4[i] = val;
}

// ---------------------------------------------------------------------------
// Kernel 5: dense attention for the 45 selected queries per (h,b).
// 3 waves per block; wave mt owns query rows [16*mt, 16*mt+16).
// Two-pass flash softmax; GEMMs via V_WMMA_F32_16X16X4_F32; K/V staged with
// double-buffered GLOBAL_LOAD_ASYNC_TO_LDS_B128 (ASYNCcnt).
//
// Layout assumptions (ISA 7.12.2):
//  A (16x4 f32):  lane m = lane&15 is row M; VGPR0/1 hold K = 2*(lane>>4)+{0,1}
//  B (4x16 f32):  lane n = lane&15 is col N; VGPR0/1 hold K = 2*(lane>>4)+{0,1}
//  C/D (16x16):   VGPR j: lanes 0-15 -> row j, lanes 16-31 -> row j+8
// ---------------------------------------------------------------------------
__global__ __launch_bounds__(96) void attn_kernel(
    const float* __restrict__ q, const float* __restrict__ k,
    const float* __restrict__ v, const int* __restrict__ m_top,
    float* __restrict__ out)
{
    __shared__ float Klds[2][16 * LDS_PITCH];   // double-buffered K block
    __shared__ float Vlds[2][16 * LDS_PITCH];   // double-buffered V block
    __shared__ float Plds[3][16 * 17];          // per-wave C->A transpose buffer

    const int hb   = blockIdx.x;
    const int tid  = threadIdx.x;
    const int wave = tid >> 5;            // M-tile 0..2
    const int lane = tid & 31;
    const int m    = lane & 15;
    const int half = lane >> 4;

    const size_t base = (size_t)hb * SEQ_L * DIM_D;
    const float* Qb = q + base;
    const uint64_t kbase = (uint64_t)(uintptr_t)(k + base);
    const uint64_t vbase = (uint64_t)(uintptr_t)(v + base);

    const uint32_t koff0 = lds_byte_off(&Klds[0][0]);
    const uint32_t koff1 = lds_byte_off(&Klds[1][0]);
    const uint32_t voff0 = lds_byte_off(&Vlds[0][0]);
    const uint32_t voff1 = lds_byte_off(&Vlds[1][0]);

    // ---- load Q tile (A-format registers), pre-scaled; pad rows -> 0 ----
    const int u    = wave * 16 + m;
    const int qrow = (u < N_TOP) ? m_top[hb * N_TOP_PAD + u] : -1;
    v2f aq[16];
    for (int c = 0; c < 16; ++c) {
        if (qrow >= 0) {
            const float* qp = Qb + (size_t)qrow * DIM_D + 4 * c + 2 * half;
            aq[c].x = qp[0] * SCALE;
            aq[c].y = qp[1] * SCALE;
        } else {
            aq[c].x = 0.0f; aq[c].y = 0.0f;
        }
    }

    float mrow[8], lrow[8];
    for (int j = 0; j < 8; ++j) { mrow[j] = -INFINITY; lrow[j] = 0.0f; }

    // ---------------- pass 1: row max + sum(exp) -------------------------
    issue_k_stage(koff0, kbase, 0, tid);                       // prefetch block 0
    for (int kb = 0; kb < KB_COUNT; ++kb) {
        const int cur = kb & 1;
        wait_asynccnt0();          // my loads for block kb landed
        __syncthreads();           // everyone's landed; prior reads of nxt done
        if (kb + 1 < KB_COUNT)
            issue_k_stage((kb + 1) & 1 ? koff1 : koff0, kbase, kb + 1, tid);

        v8f S = {};
        const float* Kc = &Klds[cur][0];
        for (int c = 0; c < 16; ++c) {
            const int kd = 4 * c + 2 * half;
            v2f b;
            b.x = Kc[m * LDS_PITCH + kd];
            b.y = Kc[m * LDS_PITCH + kd + 1];
            S = __builtin_amdgcn_wmma_f32_16x16x4_f32(false, aq[c], false, b,
                                                      (short)0, S, false, false);
        }
        for (int j = 0; j < 8; ++j) {
            float s = S[j];
            float t = s;   // half-wave (16-lane) row reductions
            t = fmaxf(t, __shfl_xor(t, 1, 32));
            t = fmaxf(t, __shfl_xor(t, 2, 32));
            t = fmaxf(t, __shfl_xor(t, 4, 32));
            t = fmaxf(t, __shfl_xor(t, 8, 32));
            const float mnew = fmaxf(mrow[j], t);
            float e = __expf(s - mnew);
            e += __shfl_xor(e, 1, 32);
            e += __shfl_xor(e, 2, 32);
            e += __shfl_xor(e, 4, 32);
            e += __shfl_xor(e, 8, 32);
            lrow[j] = lrow[j] * __expf(mrow[j] - mnew) + e;
            mrow[j] = mnew;
        }
    }

    // ---------------- pass 2: O = exp(S - m) @ V -------------------------
    v8f O[4];
    for (int nt = 0; nt < 4; ++nt) O[nt] = (v8f){};

    issue_kv_stage(koff0, kbase, voff0, vbase, 0, tid);        // prefetch block 0
    for (int kb = 0; kb < KB_COUNT; ++kb) {
        const int cur = kb & 1;
        wait_asynccnt0();
        __syncthreads();
        if (kb + 1 < KB_COUNT) {
            const int nxt = (kb + 1) & 1;
            issue_kv_stage(nxt ? koff1 : koff0, kbase,
                           nxt ? voff1 : voff0, vbase, kb + 1, tid);
        }

        v8f S = {};
        const float* Kc = &Klds[cur][0];
        const float* Vc = &Vlds[cur][0];
        for (int c = 0; c < 16; ++c) {
            const int kd = 4 * c + 2 * half;
            v2f b;
            b.x = Kc[m * LDS_PITCH + kd];
            b.y = Kc[m * LDS_PITCH + kd + 1];
            S = __builtin_amdgcn_wmma_f32_16x16x4_f32(false, aq[c], false, b,
                                                      (short)0, S, false, false);
        }
        // P = exp(S - m) ; C-layout -> LDS (row j / j+8 per half)
        for (int j = 0; j < 8; ++j) {
            Plds[wave][(j + 8 * half) * 17 + m] = __expf(S[j] - mrow[j]);
        }
        // Same-wave LDS store->load is in-order (DScnt, per ISA §7.3).
        for (int nt = 0; nt < 4; ++nt) {
            for (int kc = 0; kc < 4; ++kc) {
                const int kd = 4 * kc + 2 * half;
                v2f a, b;
                a.x = Plds[wave][m * 17 + kd];
                a.y = Plds[wave][m * 17 + kd + 1];
                b.x = Vc[kd * LDS_PITCH + 16 * nt + m];
                b.y = Vc[(kd + 1) * LDS_PITCH + 16 * nt + m];
                O[nt] = __builtin_amdgcn_wmma_f32_16x16x4_f32(false, a, false, b,
                                                              (short)0, O[nt], false, false);
            }
        }
    }

    // ---- scatter the 45 real rows: out[hb, m_top[u], :] = O / l ---------
    for (int j = 0; j < 8; ++j) {
        const int row = j + 8 * half;           // C-layout row for this lane
        const int uu  = wave * 16 + row;
        if (uu < N_TOP) {
            const int   grow = m_top[hb * N_TOP_PAD + uu];
            const float inv  = 1.0f / lrow[j];
            for (int nt = 0; nt < 4; ++nt) {
                out[base + (size_t)grow * DIM_D + 16 * nt + m] = O[nt][j] * inv;
            }
        }
    }
}

// ---------------------------------------------------------------------------
// Launch: inputs = {q, k, v, sample_idx}; output = ctx [H,B,L,D] fp32.
// Workspace layout: M (4 MB) | m_top (48 KB in a 64 KB region) | vmean (64 KB region)
// ---------------------------------------------------------------------------
extern "C" void kernel_launch(void* const* d_in, const int* in_sizes, int n_in,
                              void* d_out, int out_size, void* d_ws, size_t ws_size,
                              hipStream_t stream) {
    const float* q    = (const float*)d_in[0];
    const float* k    = (const float*)d_in[1];
    const float* v    = (const float*)d_in[2];
    const int*   sidx = (const int*)d_in[3];
    float*       out  = (float*)d_out;

    char* ws = (char*)d_ws;
    float* Mws   = (float*)ws;                                   // 256*4096 f32 = 4 MB
    int*   m_top = (int*)(ws + (size_t)4 * 1024 * 1024);         // 256*48 i32
    float* vmean = (float*)(ws + (size_t)4 * 1024 * 1024 + 64 * 1024); // 256*64 f32

    // K1: 1M queries, one wave each, 8 waves/block
    {
        const int total_waves = HB_TOTAL * SEQ_L;
        sample_m_kernel<<<total_waves / 8, 256, 0, stream>>>(q, k, sidx, Mws);
    }
    // K2: top-45 per (h,b)
    topk_kernel<<<HB_TOTAL, 256, 0, stream>>>(Mws, m_top);
    // K3: mean of V per (h,b)
    vmean_kernel<<<HB_TOTAL, 256, 0, stream>>>(v, vmean);
    // K4: broadcast fill of the whole output (16M float4)
    {
        const int n4 = HB_TOTAL * SEQ_L * (DIM_D / 4);           // 16,777,216
        fill_kernel<<<n4 / 256, 256, 0, stream>>>(vmean, (float4*)out);
    }
    // K5: WMMA attention + scatter of attended rows
    attn_kernel<<<HB_TOTAL, 96, 0, stream>>>(q, k, v, m_top, out);
}